// StageALoss_32538672234942
// MI455X (gfx1250) — compile-verified
//
#include <hip/hip_runtime.h>
#include <hip/hip_bf16.h>
#include <float.h>
#include <limits.h>
#include <math.h>

// ---------------------------------------------------------------------------
// StageA detection loss for MI455X (gfx1250, wave32).
//   - 384 workgroups = (B=128) x (3 pyramid levels), 256 threads (8 waves).
//   - Streams only the det maps (~55 MB); qual maps are gathered at 1 cell.
//   - Wave-level sums via V_WMMA_F32_16X16X4_F32 (B = ones -> row sums),
//     cross-wave combine in LDS, global float atomics into workspace.
// ---------------------------------------------------------------------------

#define NB 128
#define CQ 4
#define THREADS 256
#define NWAVES (THREADS / 32)

#define FOCAL_ALPHA 0.25f
#define FOCAL_GAMMA_IS_2 1
#define EPS 1e-6f
#define PI_F 3.14159265358979323846f

typedef float v2f __attribute__((ext_vector_type(2)));
typedef float v8f __attribute__((ext_vector_type(8)));

// Full 32-lane sum using the f32 WMMA as a reduction engine.
// A: VGPR0 = v, VGPR1 = 0  =>  A[m][0]=v(lane m), A[m][2]=v(lane m+16).
// B: all ones (4x16)       =>  D[m][n] = v(m) + v(m+16).
// C/D layout: lanes 0-15 hold rows 0..7 in VGPR0..7, lanes 16-31 rows 8..15.
// Sum of the 8 D regs + shfl_xor(16) = full wave sum in every lane.
__device__ __forceinline__ float wave_sum32(float v) {
#if __has_builtin(__builtin_amdgcn_wmma_f32_16x16x4_f32)
    v2f a;  a[0] = v;    a[1] = 0.0f;
    v2f bo; bo[0] = 1.0f; bo[1] = 1.0f;
    v8f c = {};
    v8f d = __builtin_amdgcn_wmma_f32_16x16x4_f32(
        /*neg_a=*/false, a, /*neg_b=*/false, bo,
        /*c_mod=*/(short)0, c, /*reuse_a=*/false, /*reuse_b=*/false);
    float s = ((d[0] + d[1]) + (d[2] + d[3])) + ((d[4] + d[5]) + (d[6] + d[7]));
    s += __shfl_xor(s, 16, 32);
    return s;
#else
    #pragma unroll
    for (int off = 16; off; off >>= 1) v += __shfl_xor(v, off, 32);
    return v;
#endif
}

__device__ __forceinline__ float log_sigmoid_f(float x) {
    // stable: min(x,0) - log1p(exp(-|x|))
    return fminf(x, 0.0f) - log1pf(expf(-fabsf(x)));
}

__global__ void __launch_bounds__(THREADS)
stage_a_kernel(const float* __restrict__ det8,  const float* __restrict__ det16,
               const float* __restrict__ det32, const float* __restrict__ qual8,
               const float* __restrict__ qual16, const float* __restrict__ qual32,
               const float* __restrict__ gt_bboxes, const float* __restrict__ gt_quality,
               float* __restrict__ acc /* [0]=conf [1]=bbox [2]=qual */,
               int* __restrict__ npos_acc)
{
    const int bid = blockIdx.x;
    const int b   = bid & (NB - 1);
    const int lvl = bid >> 7;

    int H; float stride; const float* det; const float* qual;
    if (lvl == 0)      { H = 128; stride =  8.0f; det = det8;  qual = qual8;  }
    else if (lvl == 1) { H =  64; stride = 16.0f; det = det16; qual = qual16; }
    else               { H =  32; stride = 32.0f; det = det32; qual = qual32; }
    const int HW = H * H;

    const float x1 = gt_bboxes[b * 4 + 0];
    const float y1 = gt_bboxes[b * 4 + 1];
    const float x2 = gt_bboxes[b * 4 + 2];
    const float y2 = gt_bboxes[b * 4 + 3];
    const bool  valid = (x1 >= 0.0f);
    const float bcx = 0.5f * (x1 + x2);
    const float bcy = 0.5f * (y1 + y2);

    const float* __restrict__ p0 = det + ((size_t)b * 5 + 0) * (size_t)HW;
    const float* __restrict__ p1 = det + ((size_t)b * 5 + 1) * (size_t)HW;
    const float* __restrict__ p2 = det + ((size_t)b * 5 + 2) * (size_t)HW;
    const float* __restrict__ p3 = det + ((size_t)b * 5 + 3) * (size_t)HW;
    const float* __restrict__ p4 = det + ((size_t)b * 5 + 4) * (size_t)HW;

    float fsum = 0.0f;      // sum(focal * valid)
    float csum = 0.0f;      // sum(ciou * t)
    float nsum = 0.0f;      // npos (as float; exact up to 2^24)
    float dmin = FLT_MAX;   // min dist^2 among in-box cells
    int   imin = INT_MAX;   // flat argmin (first-index tie-break)

    for (int i = threadIdx.x; i < HW; i += THREADS) {
        __builtin_prefetch(p0 + i + THREADS, 0, 1);  // global_prefetch_b8

        const float z  = p0[i];
        const int   gy = i / H;
        const int   gx = i - gy * H;
        const float cx = ((float)gx + 0.5f) * stride;
        const float cy = ((float)gy + 0.5f) * stride;

        const bool inb = valid && (cx > x1) && (cx < x2) && (cy > y1) && (cy < y2);
        const float t  = inb ? 1.0f : 0.0f;

        // focal (computed for every cell, masked by valid)
        const float ce = fmaxf(z, 0.0f) - z * t + log1pf(expf(-fabsf(z)));
        const float p  = 1.0f / (1.0f + expf(-z));
        const float pt = inb ? p : (1.0f - p);
        const float at = inb ? FOCAL_ALPHA : (1.0f - FOCAL_ALPHA);
        const float om = 1.0f - pt;
        const float focal = at * om * om * ce;
        fsum += valid ? focal : 0.0f;

        if (inb) {
            // CIoU term (only contributes where t == 1)
            const float pl  = expf(fminf(fmaxf(p1[i], -6.0f), 6.0f)) * stride;
            const float pt_ = expf(fminf(fmaxf(p2[i], -6.0f), 6.0f)) * stride;
            const float pr  = expf(fminf(fmaxf(p3[i], -6.0f), 6.0f)) * stride;
            const float pb  = expf(fminf(fmaxf(p4[i], -6.0f), 6.0f)) * stride;

            const float gl  = cx - x1, gt_ = cy - y1, gr = x2 - cx, gb = y2 - cy;
            const float pw  = pl + pr,  ph = pt_ + pb;
            const float gw  = gl + gr,  gh = gt_ + gb;

            const float iw    = fmaxf(fminf(pl, gl) + fminf(pr, gr), 0.0f);
            const float ih    = fmaxf(fminf(pt_, gt_) + fminf(pb, gb), 0.0f);
            const float inter = iw * ih;
            const float uni   = pw * ph + gw * gh - inter + EPS;
            const float iou   = inter / uni;

            const float cw   = fmaxf(pl, gl) + fmaxf(pr, gr);
            const float chh  = fmaxf(pt_, gt_) + fmaxf(pb, gb);
            const float c2   = cw * cw + chh * chh + EPS;
            const float dcx  = 0.5f * (pr - pl) - 0.5f * (gr - gl);
            const float dcy  = 0.5f * (pb - pt_) - 0.5f * (gb - gt_);
            const float rho2 = dcx * dcx + dcy * dcy;

            const float da = atanf(gw / (gh + EPS)) - atanf(pw / (ph + EPS));
            const float v4 = (4.0f / (PI_F * PI_F)) * da * da;
            const float alpha_v = v4 / (1.0f - iou + v4 + EPS);  // stop_gradient: fwd value
            const float ciou = 1.0f - iou + rho2 / c2 + alpha_v * v4;

            csum += ciou;
            nsum += 1.0f;

            const float ddx = cx - bcx, ddy = cy - bcy;
            const float d2  = ddx * ddx + ddy * ddy;
            if (d2 < dmin || (d2 == dmin && i < imin)) { dmin = d2; imin = i; }
        }
    }

    // ---- wave reductions: sums via WMMA, argmin via shuffle tree -----------
    fsum = wave_sum32(fsum);
    csum = wave_sum32(csum);
    nsum = wave_sum32(nsum);
    #pragma unroll
    for (int off = 16; off; off >>= 1) {
        const float od = __shfl_xor(dmin, off, 32);
        const int   oi = __shfl_xor(imin, off, 32);
        if (od < dmin || (od == dmin && oi < imin)) { dmin = od; imin = oi; }
    }

    __shared__ float sF[NWAVES], sC[NWAVES], sN[NWAVES], sD[NWAVES];
    __shared__ int   sI[NWAVES];
    const int wave = threadIdx.x >> 5;
    const int lane = threadIdx.x & 31;
    if (lane == 0) { sF[wave] = fsum; sC[wave] = csum; sN[wave] = nsum;
                     sD[wave] = dmin; sI[wave] = imin; }
    __syncthreads();

    if (threadIdx.x == 0) {
        float F = 0.0f, C = 0.0f, N = 0.0f, D = FLT_MAX; int I = INT_MAX;
        #pragma unroll
        for (int w = 0; w < NWAVES; ++w) {
            F += sF[w]; C += sC[w]; N += sN[w];
            if (sD[w] < D || (sD[w] == D && sI[w] < I)) { D = sD[w]; I = sI[w]; }
        }
        const int npos = (int)(N + 0.5f);

        // conf: (focal*valid).mean() over (B,H,W) -> divide this image's sum
        atomicAdd(&acc[0], F / ((float)NB * (float)HW));

        if (npos > 0) {
            atomicAdd(&acc[1], C / (float)npos);   // ciou_mean for this (b,lvl)
            atomicAdd(npos_acc, 1);                // has_pos count

            // quality BCE at the argmin cell (4-channel gather)
            const int idx = (I == INT_MAX) ? 0 : I;
            const float* __restrict__ q = qual + (size_t)b * CQ * (size_t)HW;
            float bce = 0.0f;
            #pragma unroll
            for (int ch = 0; ch < CQ; ++ch) {
                const float x  = q[(size_t)ch * HW + idx];
                const float qt = fminf(fmaxf(gt_quality[b * CQ + ch], 0.05f), 0.95f);
                bce += -(qt * log_sigmoid_f(x) + (1.0f - qt) * log_sigmoid_f(-x));
            }
            atomicAdd(&acc[2], bce * (1.0f / (float)CQ));
        }
    }
}

__global__ void init_ws_kernel(float* __restrict__ acc, int* __restrict__ npos_acc) {
    if (threadIdx.x == 0) { acc[0] = 0.0f; acc[1] = 0.0f; acc[2] = 0.0f; *npos_acc = 0; }
}

__global__ void finalize_kernel(const float* __restrict__ acc,
                                const int* __restrict__ npos_acc,
                                float* __restrict__ out) {
    if (threadIdx.x == 0) {
        const float conf = acc[0];
        float bbox = acc[1];
        float qual = acc[2];
        const int n = *npos_acc;
        if (n > 0) {
            const float denom = (float)n;
            bbox /= denom;
            qual /= denom;
        }
        out[0] = conf + 5.0f * bbox + 0.05f * qual;  // total
        out[1] = conf;
        out[2] = bbox;
        out[3] = qual;
    }
}

extern "C" void kernel_launch(void* const* d_in, const int* in_sizes, int n_in,
                              void* d_out, int out_size, void* d_ws, size_t ws_size,
                              hipStream_t stream) {
    const float* det8   = (const float*)d_in[0];
    const float* det16  = (const float*)d_in[1];
    const float* det32  = (const float*)d_in[2];
    const float* qual8  = (const float*)d_in[3];
    const float* qual16 = (const float*)d_in[4];
    const float* qual32 = (const float*)d_in[5];
    const float* gt_bboxes  = (const float*)d_in[6];
    const float* gt_quality = (const float*)d_in[7];

    float* acc      = (float*)d_ws;
    int*   npos_acc = (int*)((char*)d_ws + 3 * sizeof(float));

    init_ws_kernel<<<1, 32, 0, stream>>>(acc, npos_acc);
    stage_a_kernel<<<NB * 3, THREADS, 0, stream>>>(det8, det16, det32,
                                                   qual8, qual16, qual32,
                                                   gt_bboxes, gt_quality,
                                                   acc, npos_acc);
    finalize_kernel<<<1, 32, 0, stream>>>(acc, npos_acc, (float*)d_out);
}